// MeshLoss_36292473651892
// MI455X (gfx1250) — compile-verified
//
#include <hip/hip_runtime.h>

typedef __attribute__((ext_vector_type(2))) float v2f;
typedef __attribute__((ext_vector_type(8))) float v8f;

#define BIGF   1e10f
#define NB     4        // batches
#define NV     4096     // mesh-top points per batch (64*64)
#define NM     8192     // cloud points per batch
#define VB_STR 393216   // 3*64*32*64
#define VC_STR 131072   // 64*32*64

// vertices[b, c, x2, 31, x4] with n = x2*64 + x4, then (val - 0.5) * 2
__device__ __forceinline__ float vload(const float* __restrict__ V, int b, int c, int n) {
    int x2 = n >> 6, x4 = n & 63;
    return (V[b * VB_STR + c * VC_STR + x2 * 2048 + 1984 + x4] - 0.5f) * 2.0f;
}
__device__ __forceinline__ float pload(const float* __restrict__ P, int b, int c, int m) {
    return P[b * 24576 + c * 8192 + m];
}

__global__ void k_zero(float* rowSum) {
    if (threadIdx.x < NB) rowSum[threadIdx.x] = 0.0f;
}

// Build lane-ready WMMA fragments for cloud points:
//   pfrag[b][tile*32+lane] = lane<16 ? (x,y) : (z,0)   (A- and B-fragment compatible)
//   adjdup[..]             = valid ? ||p||^2 : BIG     (duplicated across lane halves)
__global__ void k_prep_p(const float* __restrict__ Pc, float2* __restrict__ pfrag,
                         float* __restrict__ adjdup) {
    int idx  = (int)(blockIdx.x * blockDim.x + threadIdx.x);  // NB*2*NM = 65536
    int lane = idx & 31, half = lane >> 4, l = lane & 15;
    int tile = (idx >> 5) & 511;
    int b    = idx >> 14;
    int j    = tile * 16 + l;
    float px = pload(Pc, b, 0, j);
    float py = pload(Pc, b, 1, j);
    float pz = pload(Pc, b, 2, j);
    float2 f;
    f.x = half ? pz : px;
    f.y = half ? 0.0f : py;
    float np = px * px + py * py + pz * pz;
    bool valid = (px != 0.0f) || (py != 0.0f) || (pz != 0.0f);
    pfrag[idx]  = f;
    adjdup[idx] = valid ? np : BIGF;
}

// Same for transformed mesh-top points: vfrag + nvdup (= ||v||^2, no masking).
__global__ void k_prep_v(const float* __restrict__ Vtx, float2* __restrict__ vfrag,
                         float* __restrict__ nvdup) {
    int idx  = (int)(blockIdx.x * blockDim.x + threadIdx.x);  // NB*2*NV = 32768
    int lane = idx & 31, half = lane >> 4, l = lane & 15;
    int tile = (idx >> 5) & 255;
    int b    = idx >> 13;
    int i    = tile * 16 + l;
    float vx = vload(Vtx, b, 0, i);
    float vy = vload(Vtx, b, 1, i);
    float vz = vload(Vtx, b, 2, i);
    float2 f;
    f.x = half ? vz : vx;
    f.y = half ? 0.0f : vy;
    vfrag[idx] = f;
    nvdup[idx] = vx * vx + vy * vy + vz * vz;
}

// Each wave owns 32 v-rows (2 A fragments); loops over all 512 point tiles.
// Hot loop: 1x b64 + 1x b32 load, 2x WMMA, packed add/fma/min.
__global__ void k_rowmin(const float2* __restrict__ pfrag, const float* __restrict__ adjdup,
                         const float2* __restrict__ vfrag, const float* __restrict__ nvdup,
                         float* __restrict__ rowSum) {
    int wave = (int)((blockIdx.x * blockDim.x + threadIdx.x) >> 5);
    int lane = (int)(threadIdx.x & 31);
    int b    = wave >> 7;            // 128 row-blocks (32 rows) per batch
    int rb   = wave & 127;
    int half = lane >> 4;
    int l    = lane & 15;
    int t0   = rb * 2;

    const float2* vf = vfrag + b * (2 * NV);
    const float*  nd = nvdup + b * (2 * NV);
    float2 a0 = vf[t0 * 32 + lane];
    float2 a1 = vf[(t0 + 1) * 32 + lane];
    v2f A0; A0.x = a0.x; A0.y = a0.y;
    v2f A1; A1.x = a1.x; A1.y = a1.y;

    float nv0[8], nv1[8];
#pragma unroll
    for (int r = 0; r < 8; ++r) {
        nv0[r] = nd[t0 * 32 + r + 8 * half];
        nv1[r] = nd[(t0 + 1) * 32 + r + 8 * half];
    }

    v8f acc0, acc1;
#pragma unroll
    for (int r = 0; r < 8; ++r) { acc0[r] = BIGF; acc1[r] = BIGF; }

    const float2* pf = pfrag + b * (2 * NM);
    const float*  ad = adjdup + b * (2 * NM);
    for (int t = 0; t < NM / 16; ++t) {
        int base = t * 32 + lane;
        float2 bb = pf[base];
        float  av = ad[base];
        __builtin_prefetch(&pf[base + 256], 0, 0);   // 8 tiles ahead
        v2f Bf; Bf.x = bb.x; Bf.y = bb.y;
        v8f C = {};
        v8f d0 = __builtin_amdgcn_wmma_f32_16x16x4_f32(
            false, A0, false, Bf, (short)0, C, false, false);
        v8f d1 = __builtin_amdgcn_wmma_f32_16x16x4_f32(
            false, A1, false, Bf, (short)0, C, false, false);
#pragma unroll
        for (int r = 0; r < 8; ++r) {
            acc0[r] = fminf(acc0[r], fmaf(d0[r], -2.0f, nv0[r] + av));
            acc1[r] = fminf(acc1[r], fmaf(d1[r], -2.0f, nv1[r] + av));
        }
    }

    // min across the 16 lanes of each half-group (columns of the D tiles)
#pragma unroll
    for (int mxor = 1; mxor <= 8; mxor <<= 1) {
#pragma unroll
        for (int r = 0; r < 8; ++r) {
            acc0[r] = fminf(acc0[r], __shfl_xor(acc0[r], mxor, 32));
            acc1[r] = fminf(acc1[r], __shfl_xor(acc1[r], mxor, 32));
        }
    }
    float s = 0.0f;
#pragma unroll
    for (int r = 0; r < 8; ++r) s += acc0[r] + acc1[r];
    if (l == 0) atomicAdd(&rowSum[b], s);
}

// Each wave owns 32 cloud points (2 A fragments); loops over all 256 v tiles.
// Masked points compute with adj=BIG -> wrong colMin, but finish multiplies by mask=0.
__global__ void k_colmin(const float2* __restrict__ pfrag, const float* __restrict__ adjdup,
                         const float2* __restrict__ vfrag, const float* __restrict__ nvdup,
                         float* __restrict__ colMin) {
    int wave = (int)((blockIdx.x * blockDim.x + threadIdx.x) >> 5);
    int lane = (int)(threadIdx.x & 31);
    int b    = wave >> 8;            // 256 col-blocks (32 points) per batch
    int cb   = wave & 255;
    int half = lane >> 4;
    int l    = lane & 15;
    int t0   = cb * 2;

    const float2* pf = pfrag + b * (2 * NM);
    const float*  ad = adjdup + b * (2 * NM);
    float2 a0 = pf[t0 * 32 + lane];
    float2 a1 = pf[(t0 + 1) * 32 + lane];
    v2f A0; A0.x = a0.x; A0.y = a0.y;
    v2f A1; A1.x = a1.x; A1.y = a1.y;

    float np0[8], np1[8];
#pragma unroll
    for (int r = 0; r < 8; ++r) {
        np0[r] = ad[t0 * 32 + r + 8 * half];
        np1[r] = ad[(t0 + 1) * 32 + r + 8 * half];
    }

    v8f acc0, acc1;
#pragma unroll
    for (int r = 0; r < 8; ++r) { acc0[r] = 3.0e38f; acc1[r] = 3.0e38f; }

    const float2* vf = vfrag + b * (2 * NV);
    const float*  nd = nvdup + b * (2 * NV);
    for (int t = 0; t < NV / 16; ++t) {
        int base = t * 32 + lane;
        float2 bb = vf[base];
        float  nv = nd[base];
        __builtin_prefetch(&vf[base + 256], 0, 0);
        v2f Bf; Bf.x = bb.x; Bf.y = bb.y;
        v8f C = {};
        v8f d0 = __builtin_amdgcn_wmma_f32_16x16x4_f32(
            false, A0, false, Bf, (short)0, C, false, false);
        v8f d1 = __builtin_amdgcn_wmma_f32_16x16x4_f32(
            false, A1, false, Bf, (short)0, C, false, false);
#pragma unroll
        for (int r = 0; r < 8; ++r) {
            acc0[r] = fminf(acc0[r], fmaf(d0[r], -2.0f, np0[r] + nv));
            acc1[r] = fminf(acc1[r], fmaf(d1[r], -2.0f, np1[r] + nv));
        }
    }

#pragma unroll
    for (int mxor = 1; mxor <= 8; mxor <<= 1) {
#pragma unroll
        for (int r = 0; r < 8; ++r) {
            acc0[r] = fminf(acc0[r], __shfl_xor(acc0[r], mxor, 32));
            acc1[r] = fminf(acc1[r], __shfl_xor(acc1[r], mxor, 32));
        }
    }
    if (l == 0) {
        int j0 = t0 * 16;
#pragma unroll
        for (int r = 0; r < 8; ++r) {
            colMin[b * NM + j0 + r + 8 * half]      = acc0[r];
            colMin[b * NM + j0 + 16 + r + 8 * half] = acc1[r];
        }
    }
}

// loss = mean_b[ rowSum[b]/NV + sum(mask*colMin)/max(n_valid,1) ]
__global__ void k_finish(const float* __restrict__ Pc, const float* __restrict__ colMin,
                         const float* __restrict__ rowSum, float* __restrict__ out) {
    __shared__ float sh1[256];
    __shared__ float sh2[256];
    int tid = (int)threadIdx.x;
    float loss = 0.0f;
    for (int b = 0; b < NB; ++b) {
        float cnt = 0.0f, wsum = 0.0f;
        for (int m = tid; m < NM; m += 256) {
            float px = pload(Pc, b, 0, m);
            float py = pload(Pc, b, 1, m);
            float pz = pload(Pc, b, 2, m);
            if ((px != 0.0f) || (py != 0.0f) || (pz != 0.0f)) {
                cnt += 1.0f;
                wsum += colMin[b * NM + m];
            }
        }
        sh1[tid] = cnt;
        sh2[tid] = wsum;
        __syncthreads();
        for (int s = 128; s > 0; s >>= 1) {
            if (tid < s) { sh1[tid] += sh1[tid + s]; sh2[tid] += sh2[tid + s]; }
            __syncthreads();
        }
        if (tid == 0) {
            float nvalid = fmaxf(sh1[0], 1.0f);
            loss += rowSum[b] / (float)NV + sh2[0] / nvalid;
        }
        __syncthreads();
    }
    if (tid == 0) out[0] = loss / (float)NB;
}

extern "C" void kernel_launch(void* const* d_in, const int* in_sizes, int n_in,
                              void* d_out, int out_size, void* d_ws, size_t ws_size,
                              hipStream_t stream) {
    const float* Vtx = (const float*)d_in[0];   // [4,3,64,32,64] f32
    const float* Pc  = (const float*)d_in[1];   // [4,3,8192]     f32
    float* out = (float*)d_out;                 // scalar f32

    // workspace layout (floats); all offsets multiples of 64 -> float2-aligned
    float* w       = (float*)d_ws;
    float* rowSum  = w;                              // 4     (pad to 64)
    float* colMin  = w + 64;                         // 32768
    float* pfragF  = w + 64 + 32768;                 // NB*2*NM*2 = 131072
    float* adjdup  = pfragF + 131072;                // NB*2*NM   = 65536
    float* vfragF  = adjdup + 65536;                 // NB*2*NV*2 = 65536
    float* nvdup   = vfragF + 65536;                 // NB*2*NV   = 32768
    float2* pfrag  = (float2*)pfragF;
    float2* vfrag  = (float2*)vfragF;

    k_zero<<<1, 64, 0, stream>>>(rowSum);
    k_prep_p<<<256, 256, 0, stream>>>(Pc, pfrag, adjdup);    // 65536 threads
    k_prep_v<<<128, 256, 0, stream>>>(Vtx, vfrag, nvdup);    // 32768 threads
    // 4 batches * 128 row-blocks = 512 waves -> 64 blocks of 256
    k_rowmin<<<64, 256, 0, stream>>>(pfrag, adjdup, vfrag, nvdup, rowSum);
    // 4 batches * 256 col-blocks = 1024 waves -> 128 blocks of 256
    k_colmin<<<128, 256, 0, stream>>>(pfrag, adjdup, vfrag, nvdup, colMin);
    k_finish<<<1, 256, 0, stream>>>(Pc, colMin, rowSum, out);
}